// GroupLasso_43001212567641
// MI455X (gfx1250) — compile-verified
//
#include <hip/hip_runtime.h>

// GroupLasso regularizer for MI455X (gfx1250, wave32).
//
// Reference collapses to: out = 0.12f * (sum(W*W) + sum(b*b)).
// (A segment_sum followed by a full sum over segments is just the total sum,
// so coarse_map is mathematically irrelevant.)
//
// Roofline: ~205.2 MB of f32 reads -> ~8.8 us at 23.3 TB/s HBM (input nearly
// fits the 192 MB L2, so graph replays may run at L2 bandwidth). Compute is
// ~51.3M FMAs -- negligible. Strategy: coalesced float4 (global_load_b128)
// grid-stride streams, 4x unrolled with 4 independent accumulators so four
// b128 loads are in flight per lane; 2048 blocks x 256 threads = 16K wave32
// waves for latency hiding; deterministic two-pass reduction (no atomics).
//
// Cross-lane reduction uses V_WMMA_F32_16X16X4_F32 with an all-ones B matrix:
//   A[m][0] = acc[lane m], A[m][2] = acc[lane m+16]  (f32 16x4 A layout)
//   B = ones (layout invariant)  =>  D[m][n] = acc[m] + acc[m+16] for all n.
// Lanes 0-15 hold rows M=0..7 in D's 8 VGPRs, lanes 16-31 hold M=8..15;
// 8 adds + one ds_swizzle SWAPX16 complete the wave total.

typedef __attribute__((ext_vector_type(2))) float v2f;
typedef __attribute__((ext_vector_type(8))) float v8f;

#define TPB   256     // 8 wave32 waves per block
#define NPART 2048    // pass-1 blocks / partial slots (8 KB of d_ws)

#define SWIZZLE_SWAPX16 0x401F  // group-of-32: xor_mask=0x10, and_mask=0x1f

__device__ __forceinline__ float wave_reduce_wmma(float acc) {
  v2f a; a[0] = acc;  a[1] = 0.0f;   // A fragment: partials in VGPR0, K-odd = 0
  v2f b; b[0] = 1.0f; b[1] = 1.0f;   // B = all ones (any layout)
  v8f c = {};
  // D = A x Ones + 0 : every column n holds the row sums of A.
  v8f d = __builtin_amdgcn_wmma_f32_16x16x4_f32(
      /*neg_a=*/false, a, /*neg_b=*/false, b,
      /*c_mod=*/(short)0, c, /*reuse_a=*/false, /*reuse_b=*/false);
  float s = ((d[0] + d[1]) + (d[2] + d[3])) + ((d[4] + d[5]) + (d[6] + d[7]));
  // lanes 0-15: sum of rows M=0..7; lanes 16-31: rows M=8..15. Swap halves.
  s += __int_as_float(
      __builtin_amdgcn_ds_swizzle(__float_as_int(s), SWIZZLE_SWAPX16));
  return s;  // full wave total, uniform across the wave
}

__device__ __forceinline__ float block_reduce(float acc) {
  __shared__ float wsum[TPB / 32];
  const int lane = threadIdx.x & 31;
  const int wid  = threadIdx.x >> 5;
  float s = wave_reduce_wmma(acc);   // all lanes active: EXEC == all 1s
  if (lane == 0) wsum[wid] = s;
  __syncthreads();
  float total = 0.0f;
  if (threadIdx.x == 0) {
    for (int i = 0; i < TPB / 32; ++i) total += wsum[i];
  }
  return total;  // valid in thread 0 only
}

__device__ __forceinline__ float sq4(float4 v, float acc) {
  acc = fmaf(v.x, v.x, acc);
  acc = fmaf(v.y, v.y, acc);
  acc = fmaf(v.z, v.z, acc);
  acc = fmaf(v.w, v.w, acc);
  return acc;
}

__global__ void __launch_bounds__(TPB)
gl_sumsq_partial(const float* __restrict__ w, int n4w,
                 const float* __restrict__ b, int n4b,
                 int tail_w, int tail_b,
                 float* __restrict__ partials) {
  const int gid     = (int)blockIdx.x * TPB + (int)threadIdx.x;
  const int gstride = (int)gridDim.x * TPB;

  const float4* __restrict__ w4 = (const float4*)w;
  const float4* __restrict__ b4 = (const float4*)b;

  // Weights stream: 4x unrolled, 4 independent accumulators -> four
  // global_load_b128 in flight per lane before the FMA chain consumes them.
  float a0 = 0.0f, a1 = 0.0f, a2 = 0.0f, a3 = 0.0f;
  int i = gid;
  for (; i + 3 * gstride < n4w; i += 4 * gstride) {
    float4 v0 = w4[i];
    float4 v1 = w4[i + gstride];
    float4 v2 = w4[i + 2 * gstride];
    float4 v3 = w4[i + 3 * gstride];
    a0 = sq4(v0, a0);
    a1 = sq4(v1, a1);
    a2 = sq4(v2, a2);
    a3 = sq4(v3, a3);
  }
  for (; i < n4w; i += gstride) a0 = sq4(w4[i], a0);

  // Bias stream (25K float4 total -- one short pass).
  for (int j = gid; j < n4b; j += gstride) a1 = sq4(b4[j], a1);

  float acc = (a0 + a1) + (a2 + a3);

  // Scalar tails (sizes here are multiples of 4, but stay general).
  if (blockIdx.x == 0 && threadIdx.x == 0) {
    for (int t = 0; t < tail_w; ++t) { float x = w[4 * n4w + t]; acc = fmaf(x, x, acc); }
    for (int t = 0; t < tail_b; ++t) { float x = b[4 * n4b + t]; acc = fmaf(x, x, acc); }
  }

  float t = block_reduce(acc);
  if (threadIdx.x == 0) partials[blockIdx.x] = t;  // every slot overwritten each call
}

__global__ void __launch_bounds__(TPB)
gl_finalize(const float* __restrict__ partials, int npart,
            float* __restrict__ out, float scale) {
  float acc = 0.0f;
  for (int i = threadIdx.x; i < npart; i += TPB) acc += partials[i];
  float t = block_reduce(acc);
  if (threadIdx.x == 0) out[0] = scale * t;  // overwrite, no pre-zero needed
}

extern "C" void kernel_launch(void* const* d_in, const int* in_sizes, int n_in,
                              void* d_out, int out_size, void* d_ws, size_t ws_size,
                              hipStream_t stream) {
  const float* w = (const float*)d_in[0];   // fc_weights [NUM_FINE*FEAT_DIM] f32
  const float* b = (const float*)d_in[1];   // fc_bias    [NUM_FINE]          f32
  // d_in[2] (coarse_map, int32) is mathematically irrelevant: summing all
  // segment sums equals summing all rows.

  const int nw  = in_sizes[0];
  const int nb  = in_sizes[1];
  const int n4w = nw / 4, tail_w = nw - 4 * n4w;
  const int n4b = nb / 4, tail_b = nb - 4 * n4b;

  int npart = NPART;
  if (ws_size < (size_t)npart * sizeof(float))
    npart = (int)(ws_size / sizeof(float));  // defensive; d_ws need is only 8 KB
  float* partials = (float*)d_ws;

  gl_sumsq_partial<<<npart, TPB, 0, stream>>>(w, n4w, b, n4b, tail_w, tail_b, partials);

  const float scale = 0.02f + 0.04f + 0.06f;  // root + coarse + fine gammas = 0.12
  gl_finalize<<<1, TPB, 0, stream>>>(partials, npart, (float*)d_out, scale);
}